// HeuristicDropoutWithAlternativeRound_31172872634758
// MI455X (gfx1250) — compile-verified
//
#include <hip/hip_runtime.h>
#include <hip/hip_bf16.h>
#include <stdint.h>

// ---------------------------------------------------------------------------
// HeuristicDropoutWithAlternativeRound for MI455X (gfx1250)
//
// x: [16,256,128,128] fp32. Memory-bound: ~768MB of mandatory HBM traffic
// (stats read + conv read + write) ~= 33us at 23.3 TB/s. Three kernels:
//   1) stats:   1 block / channel, b128 loads, register histogram,
//               LDS tree + V_WMMA_F32_16X16X4_F32 lane-reduction epilogue.
//   2) select:  per-batch rank of 256 scores, top-26 mask (stable tie-break).
//   3) conv:    selected channels -> 3x3 Laplace via LDS tile staged with
//               GLOBAL_LOAD_ASYNC_TO_LDS_B128 (+ s_wait_asynccnt);
//               unselected -> straight coalesced float4 copy.
// ---------------------------------------------------------------------------

#define BATCH 16
#define CHANS 256
#define HDIM  128
#define WDIM  128
#define HW    (HDIM * WDIM)          // 16384
#define NBC   (BATCH * CHANS)        // 4096
#define KSEL  26                     // round(0.1 * 256)

typedef float v2f __attribute__((ext_vector_type(2)));
typedef float v8f __attribute__((ext_vector_type(8)));

// Wave32 all-lane sum. WMMA f32 16x16x4: with A == all-ones, D[m][n] is the
// column sum of B. Putting per-lane value v in B's VGPR0 and zero in VGPR1,
// the two K-rows held by VGPR0 are (v_n, v_{n+16}) for column n, so
// D vgpr0 lane n == v_n + v_{n+16}  (rows of D are identical). Remaining 16
// partials are folded with 4 xor-shuffles. Falls back to pure shuffles if the
// builtin is unavailable.
__device__ __forceinline__ float wave_reduce_sum(float v) {
#if __has_builtin(__builtin_amdgcn_wmma_f32_16x16x4_f32)
  v2f a; a[0] = 1.0f; a[1] = 1.0f;      // A = ones (any layout of ones is ones)
  v2f b; b[0] = v;    b[1] = 0.0f;      // one K-row pair carries the data
  v8f c = {};
  v8f d = __builtin_amdgcn_wmma_f32_16x16x4_f32(
      /*neg_a=*/false, a, /*neg_b=*/false, b,
      /*c_mod=*/(short)0, c, /*reuse_a=*/false, /*reuse_b=*/false);
  float r = d[0];                       // lane n: v_n + v_{n+16} (n mod 16)
  r += __shfl_xor(r, 1, 32);
  r += __shfl_xor(r, 2, 32);
  r += __shfl_xor(r, 4, 32);
  r += __shfl_xor(r, 8, 32);
  return r;
#else
  #pragma unroll
  for (int off = 1; off <= 16; off <<= 1) v += __shfl_xor(v, off, 32);
  return v;
#endif
}

// ---------------------------------------------------------------------------
// Kernel 1: per-channel score = entropy(hist(round(tanh(x)*10))) + 2/(var+1e-7)
// ---------------------------------------------------------------------------
__global__ __launch_bounds__(256)
void stats_kernel(const float* __restrict__ x, float* __restrict__ scores) {
  const int bc = blockIdx.x;
  const int t  = threadIdx.x;
  const float4* src = reinterpret_cast<const float4*>(x + (size_t)bc * HW);

  // Histogram must stay in registers: unrolled per-bin compares (a dynamically
  // indexed array would spill to scratch -> real VMEM RMW traffic per element).
  float hist[11];
  #pragma unroll
  for (int k = 0; k < 11; ++k) hist[k] = 0.0f;
  float sum = 0.0f, sumsq = 0.0f;

  #pragma unroll 4
  for (int i = 0; i < 16; ++i) {
    if (i + 4 < 16) __builtin_prefetch(&src[t + 256 * (i + 4)], 0, 0);
    float4 v4 = src[t + 256 * i];
    const float vv[4] = {v4.x, v4.y, v4.z, v4.w};
    #pragma unroll
    for (int j = 0; j < 4; ++j) {
      const float xv = vv[j];
      sum   += xv;
      sumsq += xv * xv;
      const float q = rintf(tanhf(xv) * 10.0f);  // RTE == jnp.round
      #pragma unroll
      for (int k = 0; k < 11; ++k) hist[k] += (q == (float)k) ? 1.0f : 0.0f;
    }
  }

  // Block reduction of 13 stats: tree 256 -> 32 in LDS, then WMMA+shuffle.
  __shared__ float red[13][256];
  #pragma unroll
  for (int k = 0; k < 11; ++k) red[k][t] = hist[k];
  red[11][t] = sum;
  red[12][t] = sumsq;
  __syncthreads();
  for (int s = 128; s >= 32; s >>= 1) {
    if (t < s) {
      #pragma unroll
      for (int k = 0; k < 13; ++k) red[k][t] += red[k][t + s];
    }
    __syncthreads();
  }

  if (t < 32) {   // wave 0, EXEC all-ones -> WMMA legal
    float tot[13];
    #pragma unroll
    for (int k = 0; k < 13; ++k) tot[k] = wave_reduce_sum(red[k][t]);
    if (t == 0) {
      float N = 0.0f;
      #pragma unroll
      for (int k = 0; k < 11; ++k) N += tot[k];
      float ent = 0.0f;
      if (N > 0.0f) {
        const float invN = 1.0f / N;
        #pragma unroll
        for (int k = 0; k < 11; ++k) {
          const float h = tot[k];
          if (h > 0.0f) { const float p = h * invN; ent -= p * logf(p); }
        }
      }
      const float n   = (float)HW;
      const float var = (tot[12] - tot[11] * tot[11] / n) / (n - 1.0f);
      scores[bc] = ent + 2.0f / (var + 1e-7f);
    }
  }
}

// ---------------------------------------------------------------------------
// Kernel 2: per-batch top-26 selection. rank(c) = #{c' : s[c']>s[c]} +
// #{c' < c : s[c']==s[c]}  (matches stable argsort(-score)); sel = rank < 26.
// ---------------------------------------------------------------------------
__global__ __launch_bounds__(256)
void select_kernel(const float* __restrict__ scores, int* __restrict__ sel) {
  __shared__ float sc[CHANS];
  const int b = blockIdx.x, c = threadIdx.x;
  const float v = scores[b * CHANS + c];
  sc[c] = v;
  __syncthreads();
  int rank = 0;
  for (int j = 0; j < CHANS; ++j) {
    const float u = sc[j];
    rank += (u > v) || (u == v && j < c);
  }
  sel[b * CHANS + c] = (rank < KSEL) ? 1 : 0;
}

// ---------------------------------------------------------------------------
// Kernel 3: conv-or-copy. Tile = 16 output rows x 128 cols per block.
// Selected channels stage an 18 x (128+halo) tile in LDS using the CDNA5
// async global->LDS DMA path; unselected channels bypass LDS entirely.
// ---------------------------------------------------------------------------
__global__ __launch_bounds__(256)
void conv_kernel(const float* __restrict__ x, const int* __restrict__ sel,
                 float* __restrict__ out) {
  const int bc = blockIdx.y;
  const int ty = blockIdx.x;                 // 8 tiles of 16 rows
  const int t  = threadIdx.x;
  const int y0 = ty * 16;
  const float* src = x   + (size_t)bc * HW;
  float*       dst = out + (size_t)bc * HW;

  // LDS tile: rows y0-1 .. y0+16, interior cols at [4..131] (16B aligned),
  // left halo col 3, right halo col 132. Row stride 136 floats = 34*16B.
  __shared__ __align__(16) float tile[18][136];

  if (!sel[bc]) {
    // Pass-through: 2048 floats = 512 float4, coalesced.
    const float4* s4 = reinterpret_cast<const float4*>(src + y0 * WDIM);
    float4*       d4 = reinterpret_cast<float4*>(dst + y0 * WDIM);
    d4[t]       = s4[t];
    d4[t + 256] = s4[t + 256];
    return;
  }

  // Stage 18 rows x 32 float4 via async DMA to LDS (zero-fill padded rows).
  for (int slot = t; slot < 18 * 32; slot += 256) {
    const int r  = slot >> 5;
    const int qx = slot & 31;
    const int gy = y0 + r - 1;
    float* ldst = &tile[r][4 + qx * 4];
    if (gy >= 0 && gy < HDIM) {
      const float* gsrc = src + gy * WDIM + qx * 4;
      const uint32_t lds_off = (uint32_t)(uintptr_t)ldst;  // addr[31:0] == LDS offset
      asm volatile("global_load_async_to_lds_b128 %0, %1, off"
                   :: "v"(lds_off), "v"(gsrc) : "memory");
    } else {
      float4 z = {0.f, 0.f, 0.f, 0.f};
      *reinterpret_cast<float4*>(ldst) = z;
    }
  }
  // Zero-pad halo columns (w-edge zero padding), all 18 rows.
  if (t < 36) {
    const int r = t >> 1;
    tile[r][(t & 1) ? 132 : 3] = 0.0f;
  }
  asm volatile("s_wait_asynccnt 0" ::: "memory");  // drain this wave's async DMA
  __syncthreads();

  // Each thread: 2 rows x 4 cols. out = 8*c - sum(8 neighbors).
  const int xq = (t & 31) * 4;   // 0..124
  const int ry = t >> 5;         // 0..7
  #pragma unroll
  for (int rr = 0; rr < 2; ++rr) {
    const int oy = ry + rr * 8;          // 0..15 in tile
    const int lr = oy + 1;               // LDS row
    const float* up  = &tile[lr - 1][4 + xq];
    const float* mid = &tile[lr    ][4 + xq];
    const float* dn  = &tile[lr + 1][4 + xq];
    float4 res;
    float* rp = reinterpret_cast<float*>(&res);
    #pragma unroll
    for (int j = 0; j < 4; ++j) {
      const float c = mid[j];
      const float s8 = up[j - 1] + up[j] + up[j + 1]
                     + mid[j - 1] + mid[j + 1]
                     + dn[j - 1] + dn[j] + dn[j + 1];
      rp[j] = 8.0f * c - s8;
    }
    *reinterpret_cast<float4*>(dst + (y0 + oy) * WDIM + xq) = res;
  }
}

// ---------------------------------------------------------------------------
extern "C" void kernel_launch(void* const* d_in, const int* in_sizes, int n_in,
                              void* d_out, int out_size, void* d_ws, size_t ws_size,
                              hipStream_t stream) {
  const float* x = (const float*)d_in[0];
  float* out = (float*)d_out;
  float* scores = (float*)d_ws;                               // 4096 floats
  int*   sel    = (int*)((char*)d_ws + NBC * sizeof(float));  // 4096 ints

  stats_kernel<<<NBC, 256, 0, stream>>>(x, scores);
  select_kernel<<<BATCH, 256, 0, stream>>>(scores, sel);
  dim3 gridC(HDIM / 16, NBC);
  conv_kernel<<<gridC, 256, 0, stream>>>(x, sel, out);
}